// GraphNN_48498770707162
// MI455X (gfx1250) — compile-verified
//
#include <hip/hip_runtime.h>
#include <hip/hip_bf16.h>

typedef __attribute__((ext_vector_type(2))) float v2f;
typedef __attribute__((ext_vector_type(8))) float v8f;

#define N_NODES 50000
#define N_EDGES 800000
#define IN_DIM  128
#define HID_DIM 128
#define OUT_DIM 64

// ---------------- degree / normalization ----------------
__global__ void k_deg_init(float* deg, int n) {
    int i = blockIdx.x * blockDim.x + threadIdx.x;
    if (i < n) deg[i] = 1.0f;  // self-loop contributes 1
}

__global__ void k_deg_count(const int* __restrict__ dst, float* deg, int e) {
    int i = blockIdx.x * blockDim.x + threadIdx.x;
    if (i < e) atomicAdd(&deg[dst[i]], 1.0f);
}

__global__ void k_dis(float* deg, int n) {
    int i = blockIdx.x * blockDim.x + threadIdx.x;
    if (i < n) deg[i] = rsqrtf(deg[i]);  // deg >= 1 always (self-loops)
}

// ---------------- weight transpose: Bt[n*K + k] = B[k*N + n] ----------------
__global__ void k_transpose(const float* __restrict__ B, float* __restrict__ Bt, int K, int N) {
    int i = blockIdx.x * blockDim.x + threadIdx.x;
    if (i < K * N) {
        int k = i / N, n = i % N;
        Bt[(size_t)n * K + k] = B[i];
    }
}

// ---------------- fp32 WMMA GEMM: C[M,NCOLS] = A[M,128] @ W[128,NCOLS] ----------------
// Bt is the transposed weight [NCOLS,128]. Block = (NCOLS/16) waves; wave w owns
// columns [16w,16w+16) and caches its whole 128x16 B slice in registers (64 VGPR/lane,
// loaded as b64 pairs). Block sweeps RTILES row tiles, staging each 16x128 A tile in
// LDS via float4 (b128) loads, so weight traffic is amortized 8x.
template <int NCOLS, int RTILES>
__global__ __launch_bounds__(NCOLS * 2) void k_gemm_wmma(const float* __restrict__ A,
                                                         const float* __restrict__ Bt,
                                                         float* __restrict__ C, int M) {
    __shared__ float As[16 * 128];

    const int lane = threadIdx.x & 31;
    const int wave = threadIdx.x >> 5;
    const int m    = lane & 15;      // row within tile (A / C layout)
    const int half = lane >> 4;      // K-half selector (f32 WMMA layout)
    const int n    = wave * 16 + m;  // output column

    // Register-cache this wave's B slice: lane holds (kA, kA+1) of column n per k-step.
    v2f bf[32];
    const float* bptr = Bt + (size_t)n * 128;
#pragma unroll
    for (int t = 0; t < 32; ++t) {
        const int kA = t * 4 + half * 2;
        bf[t]        = *(const v2f*)(bptr + kA);  // contiguous -> global_load_b64
    }

    for (int rt = 0; rt < RTILES; ++rt) {
        const int rowBase = (blockIdx.x * RTILES + rt) * 16;

        __syncthreads();  // previous tile's readers done before overwrite
        for (int idx = threadIdx.x; idx < 16 * 32; idx += blockDim.x) {
            const int r = rowBase + (idx >> 5);
            float4    v = make_float4(0.f, 0.f, 0.f, 0.f);
            if (r < M) v = ((const float4*)(A + (size_t)r * 128))[idx & 31];
            ((float4*)As)[idx] = v;  // b128 global load + b128 LDS store
        }
        __syncthreads();

        v8f acc = {};
#pragma unroll
        for (int t = 0; t < 32; ++t) {
            const int kA = t * 4 + half * 2;
            v2f       a;
            a.x = As[m * 128 + kA];
            a.y = As[m * 128 + kA + 1];
            acc = __builtin_amdgcn_wmma_f32_16x16x4_f32(false, a, false, bf[t], (short)0,
                                                        acc, false, false);
        }

        // C/D layout: VGPR j -> M = j + 8*half, N = lane%16 (+16*wave)
#pragma unroll
        for (int j = 0; j < 8; ++j) {
            const int row = rowBase + j + half * 8;
            if (row < M) C[(size_t)row * NCOLS + n] = acc[j];
        }
    }
}

// ---------------- aggregation ----------------
// out[i] = H[i] * dis[i]^2  (self-loop term, also serves as the zero-init)
__global__ void k_selfloop_init(const float* __restrict__ H, const float* __restrict__ dis,
                                float* __restrict__ out, int n, int D) {
    int i = blockIdx.x * blockDim.x + threadIdx.x;
    if (i < n * D) {
        int   node = i / D;
        float s    = dis[node];
        out[i]     = H[i] * s * s;
    }
}

// one wave32 per edge: out[dst] += H[src] * dis[src]*dis[dst]
__global__ void k_edge_agg(const int* __restrict__ src, const int* __restrict__ dst,
                           const float* __restrict__ H, const float* __restrict__ dis,
                           float* __restrict__ out, int e, int D) {
    int gid  = blockIdx.x * blockDim.x + threadIdx.x;
    int eid  = gid >> 5;
    int lane = gid & 31;
    if (eid >= e) return;
    int   s    = src[eid];
    int   d    = dst[eid];
    float norm = dis[s] * dis[d];
    const float* hrow = H + (size_t)s * D;
    float*       orow = out + (size_t)d * D;
    for (int c = lane; c < D; c += 32) atomicAdd(&orow[c], hrow[c] * norm);
}

__global__ void k_bias_relu(const float* __restrict__ in, const float* __restrict__ b,
                            float* __restrict__ out, int n, int D) {
    int i = blockIdx.x * blockDim.x + threadIdx.x;
    if (i < n * D) {
        float v = in[i] + b[i % D];
        out[i]  = v > 0.0f ? v : 0.0f;
    }
}

// ---------------- final mean ----------------
__global__ void k_out_init(const float* __restrict__ b2, float* out) {
    if (threadIdx.x < OUT_DIM) out[threadIdx.x] = b2[threadIdx.x];
}

__global__ void k_colmean(const float* __restrict__ A, float* out, int n) {
    // blockDim = 64; each block reduces 256 rows, then one atomic per column
    int   c    = threadIdx.x;
    int   base = blockIdx.x * 256;
    float sum  = 0.0f;
    for (int r = 0; r < 256; ++r) {
        int i = base + r;
        if (i < n) sum += A[(size_t)i * OUT_DIM + c];
    }
    atomicAdd(&out[c], sum * (1.0f / (float)n));
}

// ---------------- launch ----------------
extern "C" void kernel_launch(void* const* d_in, const int* in_sizes, int n_in,
                              void* d_out, int out_size, void* d_ws, size_t ws_size,
                              hipStream_t stream) {
    const float* x  = (const float*)d_in[0];
    const int*   ei = (const int*)d_in[1];
    const float* W1 = (const float*)d_in[2];
    const float* b1 = (const float*)d_in[3];
    const float* W2 = (const float*)d_in[4];
    const float* b2 = (const float*)d_in[5];
    const int* src = ei;
    const int* dst = ei + N_EDGES;
    float*     out = (float*)d_out;

    // workspace carve-out (reused: H3 aliases A1's space, A2 aliases H1's space)
    char*  ws  = (char*)d_ws;
    size_t off = 0;
    auto   carve = [&](size_t bytes) {
        void* p = ws + off;
        off += (bytes + 255) & ~(size_t)255;
        return p;
    };
    float* dis  = (float*)carve((size_t)N_NODES * 4);
    float* W1t  = (float*)carve((size_t)IN_DIM * HID_DIM * 4);
    float* W2t  = (float*)carve((size_t)HID_DIM * OUT_DIM * 4);
    float* H1   = (float*)carve((size_t)N_NODES * HID_DIM * 4);  // X@W1, later relu(A1+b1)
    float* A1   = (float*)carve((size_t)N_NODES * HID_DIM * 4);  // layer-1 aggregate
    float* H3   = A1;  // layer-2 GEMM output (A1 dead by then; 12.8MB <= 25.6MB)
    float* A2   = H1;  // layer-2 aggregate   (H1 dead after GEMM2)

    const int T      = 256;
    const int RT     = 8;               // row tiles per block
    const int ROWSPB = 16 * RT;         // 128 rows per block

    // normalization coefficients dis = (1 + in-degree)^{-1/2}
    k_deg_init<<<(N_NODES + T - 1) / T, T, 0, stream>>>(dis, N_NODES);
    k_deg_count<<<(N_EDGES + T - 1) / T, T, 0, stream>>>(dst, dis, N_EDGES);
    k_dis<<<(N_NODES + T - 1) / T, T, 0, stream>>>(dis, N_NODES);

    // weight transposes (tiny)
    k_transpose<<<(IN_DIM * HID_DIM + T - 1) / T, T, 0, stream>>>(W1, W1t, IN_DIM, HID_DIM);
    k_transpose<<<(HID_DIM * OUT_DIM + T - 1) / T, T, 0, stream>>>(W2, W2t, HID_DIM, OUT_DIM);

    // layer 1
    k_gemm_wmma<HID_DIM, RT><<<(N_NODES + ROWSPB - 1) / ROWSPB, HID_DIM * 2, 0, stream>>>(
        x, W1t, H1, N_NODES);
    k_selfloop_init<<<((size_t)N_NODES * HID_DIM + T - 1) / T, T, 0, stream>>>(
        H1, dis, A1, N_NODES, HID_DIM);
    k_edge_agg<<<((size_t)N_EDGES * 32 + T - 1) / T, T, 0, stream>>>(src, dst, H1, dis, A1,
                                                                     N_EDGES, HID_DIM);
    k_bias_relu<<<((size_t)N_NODES * HID_DIM + T - 1) / T, T, 0, stream>>>(A1, b1, H1,
                                                                           N_NODES, HID_DIM);

    // layer 2
    k_gemm_wmma<OUT_DIM, RT><<<(N_NODES + ROWSPB - 1) / ROWSPB, OUT_DIM * 2, 0, stream>>>(
        H1, W2t, H3, N_NODES);
    k_selfloop_init<<<((size_t)N_NODES * OUT_DIM + T - 1) / T, T, 0, stream>>>(
        H3, dis, A2, N_NODES, OUT_DIM);
    k_edge_agg<<<((size_t)N_EDGES * 32 + T - 1) / T, T, 0, stream>>>(src, dst, H3, dis, A2,
                                                                     N_EDGES, OUT_DIM);

    // mean over nodes + b2
    k_out_init<<<1, 64, 0, stream>>>(b2, out);
    k_colmean<<<(N_NODES + 255) / 256, 64, 0, stream>>>(A2, out, N_NODES);
}